// PolygonSegmenterGCN_72739566125683
// MI455X (gfx1250) — compile-verified
//
#include <hip/hip_runtime.h>
#include <hip/hip_bf16.h>

typedef __attribute__((ext_vector_type(2))) float v2f;
typedef __attribute__((ext_vector_type(8))) float v8f;

#define HID 64

// ---------------- degree / norm precompute ----------------

__global__ void deg_init_kernel(float* __restrict__ deg, int n) {
    int i = blockIdx.x * blockDim.x + threadIdx.x;
    if (i < n) deg[i] = 1.0f;  // self-loop weight
}

__global__ void deg_accum_kernel(const int* __restrict__ ei, const float* __restrict__ ew,
                                 float* __restrict__ deg, int E) {
    int e = blockIdx.x * blockDim.x + threadIdx.x;
    if (e < E) {
        int dst = ei[E + e];
        atomicAdd(&deg[dst], ew[e]);
    }
}

__global__ void dinv_kernel(float* __restrict__ deg, int n) {
    int i = blockIdx.x * blockDim.x + threadIdx.x;
    if (i < n) deg[i] = rsqrtf(deg[i]);  // deg >= 1 always (self-loop), no guard needed
}

__global__ void norm_kernel(const int* __restrict__ ei, const float* __restrict__ ew,
                            const float* __restrict__ dinv, float* __restrict__ norm, int E) {
    int e = blockIdx.x * blockDim.x + threadIdx.x;
    if (e < E) {
        int src = ei[e];
        int dst = ei[E + e];
        norm[e] = dinv[src] * ew[e] * dinv[dst];
    }
}

// ---------------- dense transform: out[N x 64] = H[N x K] @ W[K x 64] ----------------
// Wave computes one 16x16 output tile via V_WMMA_F32_16X16X4_F32, K-loop step 4.
// blockDim = 128 (4 waves); wave w covers output cols [16w, 16w+16); block covers 16 rows.
__global__ void gemm_wmma_kernel(const float* __restrict__ H, const float* __restrict__ W,
                                 float* __restrict__ out, int K) {
    const int lane  = threadIdx.x & 31;
    const int wave  = threadIdx.x >> 5;
    const int m0    = blockIdx.x * 16;
    const int n0    = wave * 16;
    const int half  = lane >> 4;   // 0: lanes 0-15, 1: lanes 16-31
    const int idx16 = lane & 15;

    v8f acc = {};
    const float* hrow = H + (size_t)(m0 + idx16) * K;

    for (int k = 0; k < K; k += 4) {
        // A 16x4 f32 layout: lanes 0-15 row=idx16 hold K=k..k+1; lanes 16-31 hold K=k+2..k+3
        v2f a;
        a[0] = hrow[k + 2 * half + 0];
        a[1] = hrow[k + 2 * half + 1];
        // B 4x16 f32 layout (rows striped across lanes per VGPR, K split across lane halves)
        v2f b;
        b[0] = W[(size_t)(k + 2 * half + 0) * HID + n0 + idx16];
        b[1] = W[(size_t)(k + 2 * half + 1) * HID + n0 + idx16];
        // (neg_a, A, neg_b, B, c_mod, C, reuse_a, reuse_b)
        acc = __builtin_amdgcn_wmma_f32_16x16x4_f32(false, a, false, b, (short)0, acc,
                                                    false, false);
    }

    // C/D 16x16 f32 layout: VGPR v, lane -> row = v + 8*half, col = idx16
#pragma unroll
    for (int v = 0; v < 8; ++v) {
        out[(size_t)(m0 + v + 8 * half) * HID + n0 + idx16] = acc[v];
    }
}

// ---------------- propagate ----------------

// out[i][f] = dinv[i]^2 * tmp[i][f]   (self-loop term; also serves as zero-init)
__global__ void selfloop_init_kernel(const float* __restrict__ tmp, const float* __restrict__ dinv,
                                     float* __restrict__ out, int total) {
    int i = blockIdx.x * blockDim.x + threadIdx.x;
    if (i < total) {
        float d = dinv[i >> 6];
        out[i] = d * d * tmp[i];
    }
}

// one wave per edge: gather 64-float row of tmp[src], scale by norm, atomic scatter to out[dst]
__global__ void scatter_edges_kernel(const float* __restrict__ tmp, const int* __restrict__ ei,
                                     const float* __restrict__ norm, float* __restrict__ out,
                                     int E) {
    int wid = blockIdx.x * (blockDim.x >> 5) + (threadIdx.x >> 5);
    if (wid >= E) return;
    int lane = threadIdx.x & 31;
    int src = ei[wid];
    int dst = ei[E + wid];
    float nv = norm[wid];
    const float2* s = (const float2*)(tmp + (size_t)src * HID);
    float2 v = s[lane];                                   // coalesced 256B row read
    float* o = out + (size_t)dst * HID + lane * 2;
    atomicAdd(o + 0, nv * v.x);
    atomicAdd(o + 1, nv * v.y);
}

__global__ void bias_act_kernel(float* __restrict__ out, const float* __restrict__ b,
                                int total, int relu) {
    int i = blockIdx.x * blockDim.x + threadIdx.x;
    if (i < total) {
        float v = out[i] + b[i & (HID - 1)];
        out[i] = relu ? fmaxf(v, 0.0f) : v;
    }
}

// ---------------- decode: out[p] = dot(enc[a], enc[b]) over 64 dims ----------------
__global__ void decode_kernel(const float* __restrict__ enc, const int* __restrict__ eli,
                              float* __restrict__ out, int EL) {
    int wid = blockIdx.x * (blockDim.x >> 5) + (threadIdx.x >> 5);
    if (wid >= EL) return;
    int lane = threadIdx.x & 31;
    int na = eli[wid];
    int nb = eli[EL + wid];
    const float2* pa = (const float2*)(enc + (size_t)na * HID);
    const float2* pb = (const float2*)(enc + (size_t)nb * HID);
    float2 va = pa[lane];
    float2 vb = pb[lane];
    float s = va.x * vb.x + va.y * vb.y;
#pragma unroll
    for (int off = 16; off > 0; off >>= 1) s += __shfl_down(s, off, 32);
    if (lane == 0) out[wid] = s;
}

// ---------------- driver ----------------

extern "C" void kernel_launch(void* const* d_in, const int* in_sizes, int n_in,
                              void* d_out, int out_size, void* d_ws, size_t ws_size,
                              hipStream_t stream) {
    const float* x   = (const float*)d_in[0];
    const int*   ei  = (const int*)d_in[1];
    const float* ew  = (const float*)d_in[2];
    const int*   eli = (const int*)d_in[3];
    const float* Wl[6];
    const float* Bl[6];
    for (int l = 0; l < 6; ++l) {
        Wl[l] = (const float*)d_in[4 + 2 * l];
        Bl[l] = (const float*)d_in[5 + 2 * l];
    }

    const int IN = 32;
    const int N  = in_sizes[0] / IN;       // 65536
    const int E  = in_sizes[1] / 2;        // 1048576
    const int EL = in_sizes[3] / 2;        // 2097152

    // workspace layout (floats): deg/dinv | norm | bufA (gemm out) | bufB (layer out)
    float* ws   = (float*)d_ws;
    float* deg  = ws;
    float* norm = deg + N;
    float* bufA = norm + E;
    float* bufB = bufA + (size_t)N * HID;

    const int T = 256;

    // ---- GCN normalization ----
    deg_init_kernel<<<(N + T - 1) / T, T, 0, stream>>>(deg, N);
    deg_accum_kernel<<<(E + T - 1) / T, T, 0, stream>>>(ei, ew, deg, E);
    dinv_kernel<<<(N + T - 1) / T, T, 0, stream>>>(deg, N);
    norm_kernel<<<(E + T - 1) / T, T, 0, stream>>>(ei, ew, deg, norm, E);

    // ---- 6 GCN layers ----
    const int total = N * HID;
    const float* hin = x;
    int K = IN;
    for (int l = 0; l < 6; ++l) {
        gemm_wmma_kernel<<<N / 16, 128, 0, stream>>>(hin, Wl[l], bufA, K);
        selfloop_init_kernel<<<(total + T - 1) / T, T, 0, stream>>>(bufA, deg, bufB, total);
        scatter_edges_kernel<<<(E + 7) / 8, T, 0, stream>>>(bufA, ei, norm, bufB, E);
        bias_act_kernel<<<(total + T - 1) / T, T, 0, stream>>>(bufB, Bl[l], total, l < 5 ? 1 : 0);
        hin = bufB;
        K = HID;
    }

    // ---- decode ----
    decode_kernel<<<(EL + 7) / 8, T, 0, stream>>>(bufB, eli, (float*)d_out, EL);
}